// GCN_DGL_12558484373886
// MI455X (gfx1250) — compile-verified
//
#include <hip/hip_runtime.h>

typedef __attribute__((ext_vector_type(2))) float v2f;
typedef __attribute__((ext_vector_type(8))) float v8f;

#define N_NODES   100000
#define IN_DIM    128
#define HID_DIM   32
#define OUT_DIM   12

// ---------------- zero fill ----------------
__global__ void zero_f(float* __restrict__ p, int n) {
    int t = blockIdx.x * blockDim.x + threadIdx.x;
    if (t < n) p[t] = 0.0f;
}

// ---------------- degree: deg[dst[e]] += 1 ----------------
__global__ void degree_k(const int* __restrict__ dst, float* __restrict__ deg, int nedges) {
    int e = blockIdx.x * blockDim.x + threadIdx.x;
    if (e < nedges) unsafeAtomicAdd(deg + dst[e], 1.0f);
}

// ---------------- GEMM1: H[N,32] = A[N,128] @ W[128,32]  (fp32 WMMA 16x16x4) ----
// blockDim = 128 (4 waves). Each block: 2 row-tiles x 2 col-tiles.
// A-frag layout (32-bit A 16x4): lane l: m = l&15, half = l>>4; a[j] = A[m][k + 2*half + j]
// B-frag layout (32-bit B 4x16): lane l: n = l&15;              b[j] = B[k + 2*half + j][n]
// C/D (32-bit 16x16): vgpr v, lane l: m = v + 8*half, n = l&15
__global__ void gemm1_wmma(const float* __restrict__ A,
                           const float* __restrict__ W,
                           float* __restrict__ H, int nrows) {
    int wave = threadIdx.x >> 5;
    int lane = threadIdx.x & 31;
    int half = lane >> 4;
    int ml   = lane & 15;

    int row_tile = blockIdx.x * 2 + (wave >> 1);
    int col0     = (wave & 1) * 16;
    int row0     = row_tile * 16;
    if (row0 >= nrows) return;   // wave-uniform

    v8f c = {};
    const float* arow = A + (size_t)(row0 + ml) * IN_DIM;
    for (int k = 0; k < IN_DIM; k += 4) {
        int ka = k + 2 * half;
        v2f a; a.x = arow[ka];                     a.y = arow[ka + 1];
        v2f b; b.x = W[ka * HID_DIM + col0 + ml];  b.y = W[(ka + 1) * HID_DIM + col0 + ml];
        c = __builtin_amdgcn_wmma_f32_16x16x4_f32(false, a, false, b, (short)0, c, false, false);
    }
    float* hrow = H + (size_t)row0 * HID_DIM + col0;
#pragma unroll
    for (int v = 0; v < 8; ++v) {
        int m = v + 8 * half;
        hrow[(size_t)m * HID_DIM + ml] = c[v];
    }
}

// ---------------- GEMM2: H2[N,12] = X[N,32] @ W2[32,12] (N padded to 16) -------
// blockDim = 128 (4 waves), one 16-row tile per wave, single padded col-tile.
__global__ void gemm2_wmma(const float* __restrict__ X,
                           const float* __restrict__ W2,
                           float* __restrict__ H2, int nrows) {
    int wave = threadIdx.x >> 5;
    int lane = threadIdx.x & 31;
    int half = lane >> 4;
    int ml   = lane & 15;

    int row_tile = blockIdx.x * 4 + wave;
    int row0     = row_tile * 16;
    if (row0 >= nrows) return;   // wave-uniform

    bool ncol = (ml < OUT_DIM);
    v8f c = {};
    const float* xrow = X + (size_t)(row0 + ml) * HID_DIM;
    for (int k = 0; k < HID_DIM; k += 4) {
        int ka = k + 2 * half;
        v2f a; a.x = xrow[ka]; a.y = xrow[ka + 1];
        v2f b;
        b.x = ncol ? W2[ka * OUT_DIM + ml]       : 0.0f;
        b.y = ncol ? W2[(ka + 1) * OUT_DIM + ml] : 0.0f;
        c = __builtin_amdgcn_wmma_f32_16x16x4_f32(false, a, false, b, (short)0, c, false, false);
    }
    if (ncol) {
        float* hrow = H2 + (size_t)row0 * OUT_DIM;
#pragma unroll
        for (int v = 0; v < 8; ++v) {
            int m = v + 8 * half;
            hrow[(size_t)m * OUT_DIM + ml] = c[v];
        }
    }
}

// ---------------- scatter1: agg[dst[e], f] += h[src[e], f], f in [0,32) -------
// one wave per edge, one lane per feature: coalesced 128B gather per edge.
__global__ void scatter32(const int* __restrict__ src, const int* __restrict__ dst,
                          const float* __restrict__ h, float* __restrict__ agg, int nedges) {
    int e = blockIdx.x * 8 + (threadIdx.x >> 5);
    if (e >= nedges) return;
    int f = threadIdx.x & 31;
    int s = src[e], d = dst[e];
    float v = h[(size_t)s * HID_DIM + f];
    unsafeAtomicAdd(agg + (size_t)d * HID_DIM + f, v);
}

// ---------------- scatter2: agg[dst[e], f] += h[src[e], f], f in [0,12) -------
__global__ void scatter12(const int* __restrict__ src, const int* __restrict__ dst,
                          const float* __restrict__ h, float* __restrict__ agg, int nedges) {
    int t = blockIdx.x * blockDim.x + threadIdx.x;
    int e = t >> 4;
    int f = t & 15;
    if (e >= nedges || f >= OUT_DIM) return;
    int s = src[e], d = dst[e];
    float v = h[(size_t)s * OUT_DIM + f];
    unsafeAtomicAdd(agg + (size_t)d * OUT_DIM + f, v);
}

// ---------------- finalize: x = relu(agg/deg + bias), width W ----------------
__global__ void finalize_k(const float* __restrict__ agg, const float* __restrict__ deg,
                           const float* __restrict__ bias, float* __restrict__ x,
                           int n, int width) {
    int t = blockIdx.x * blockDim.x + threadIdx.x;
    if (t >= n * width) return;
    int i = t / width;
    int j = t - i * width;
    float dg = deg[i];
    dg = dg < 1.0f ? 1.0f : dg;
    float v = agg[t] / dg + bias[j];
    x[t] = v > 0.0f ? v : 0.0f;
}

extern "C" void kernel_launch(void* const* d_in, const int* in_sizes, int n_in,
                              void* d_out, int out_size, void* d_ws, size_t ws_size,
                              hipStream_t stream) {
    const float* features = (const float*)d_in[0];
    const int*   src      = (const int*)d_in[1];
    const int*   dst      = (const int*)d_in[2];
    const float* W1       = (const float*)d_in[3];
    const float* b1       = (const float*)d_in[4];
    const float* W2       = (const float*)d_in[5];
    const float* b2       = (const float*)d_in[6];
    float* out = (float*)d_out;

    const int nedges = in_sizes[1];
    const int n      = N_NODES;

    // workspace layout (floats)
    float* ws   = (float*)d_ws;
    float* deg  = ws;                               // 100000
    float* h1   = ws + 100000;                      // 3.2M  (reused as x2h: 1.2M)
    float* agg1 = ws + 100000 + 3200000;            // 3.2M  (reused in-place as x1)
    float* agg2 = ws + 100000 + 3200000 + 3200000;  // 1.2M
    float* x1   = agg1;   // finalize1 is elementwise in-place
    float* x2h  = h1;     // gemm2 output reuses h1 region

    const int B = 256;

    // zero accumulators
    zero_f<<<(n + B - 1) / B, B, 0, stream>>>(deg, n);
    zero_f<<<(n * HID_DIM + B - 1) / B, B, 0, stream>>>(agg1, n * HID_DIM);
    zero_f<<<(n * OUT_DIM + B - 1) / B, B, 0, stream>>>(agg2, n * OUT_DIM);

    // in-degree
    degree_k<<<(nedges + B - 1) / B, B, 0, stream>>>(dst, deg, nedges);

    // layer 1
    // 100000 rows -> 6250 row-tiles of 16; 2 row-tiles x 2 col-tiles per block
    gemm1_wmma<<<6250 / 2, 128, 0, stream>>>(features, W1, h1, n);
    scatter32<<<(nedges + 7) / 8, 256, 0, stream>>>(src, dst, h1, agg1, nedges);
    finalize_k<<<(n * HID_DIM + B - 1) / B, B, 0, stream>>>(agg1, deg, b1, x1, n, HID_DIM);

    // layer 2
    gemm2_wmma<<<(6250 + 3) / 4, 128, 0, stream>>>(x1, W2, x2h, n);
    scatter12<<<(nedges * 16 + B - 1) / B, B, 0, stream>>>(src, dst, x2h, agg2, nedges);
    finalize_k<<<(n * OUT_DIM + B - 1) / B, B, 0, stream>>>(agg2, deg, b2, out, n, OUT_DIM);
}